// GMMHidden_73581379715720
// MI455X (gfx1250) — compile-verified
//
#include <hip/hip_runtime.h>
#include <hip/hip_bf16.h>
#include <stddef.h>

// ---------------------------------------------------------------------------
// Problem constants
// ---------------------------------------------------------------------------
#define C_        2
#define BS1_      9
#define NODE_     820
#define L_        3
#define K_        17
#define Q2_       289               // K*K
#define NODE1_    819               // NODE-1
#define G_        18                // C*BS1
#define ROWNODE_  2457              // NODE1*L
#define ROWS_     44226             // G*NODE1*L
#define N_XN_     752760            // G*NODE*L*K

#define SQRT2_F        1.4142135623730951f
#define INV_SQRT2PI_F  0.3989422804014327f   // 1/sqrt(2*pi)
#define EPS_F          1e-30f

// Flat offsets of the tuple outputs inside d_out (floats)
#define OFF_XN     0
#define OFF_XE     752760
#define OFF_LPN    26315388
#define OFF_LPE    27068148
#define OFF_ALPHA  39849462
#define OFF_O1     39849516
#define OFF_O2     39893742

// LDS per-row layout (stride 25 floats, odd => conflict-free column gathers)
#define RU1   0    // u1[3]
#define RI1   3    // 1/o1[3]
#define RU2   6    // u2[3]
#define RI2   9    // 1/o2[3]
#define RCF   12   // coefF[3] = alpha*inv1*inv2*C^2   (unmasked)
#define RCM   15   // coefM[3] = alpha*inv2*C          (masked: b1 == 1)
#define RMSK  18   // mask flag
#define RA1X  19   // o1_own*sqrt2
#define RA1Y  20   // u1_own
#define RA2X  21   // o2_own*sqrt2
#define RA2Y  22   // u2_own
#define RSTRIDE 25

// Gauss-Hermite nodes, n=17, ascending (numpy.polynomial.hermite.hermgauss)
__constant__ float XQ[17] = {
    -4.871345193674403f, -4.061946675875474f, -3.378932091141494f,
    -2.757762915703889f, -2.173502826666621f, -1.612924314221231f,
    -1.067648725743451f, -0.531633001342654f,  0.0f,
     0.531633001342654f,  1.067648725743451f,  1.612924314221231f,
     2.173502826666621f,  2.757762915703889f,  3.378932091141494f,
     4.061946675875474f,  4.871345193674403f };

typedef float v2f __attribute__((ext_vector_type(2)));
typedef float v8f __attribute__((ext_vector_type(8)));

// Closed form of the _ID gather table (no table in memory needed)
__device__ __forceinline__ int id_map(int n) {
    if (n < 729) {                       // id1 block
        int a = n / 81;
        int b = (n % 81) % 27;
        return 729 + a * 9 + b / 3;
    }
    if (n < 810) {                       // id2 block
        int m = n - 729;
        int p = m / 27;
        int s = m % 27;
        return 810 + 3 * p + (s % 9) / 3;
    }
    return 819;                          // tail
}

__device__ __forceinline__ void softmax3(const float* __restrict__ w, int g, float a[3]) {
    float w0 = w[g * 3 + 0], w1 = w[g * 3 + 1], w2 = w[g * 3 + 2];
    float m  = fmaxf(w0, fmaxf(w1, w2));
    float e0 = __expf(w0 - m), e1 = __expf(w1 - m), e2 = __expf(w2 - m);
    float inv = 1.0f / (e0 + e1 + e2);
    a[0] = e0 * inv; a[1] = e1 * inv; a[2] = e2 * inv;
}

// ---------------------------------------------------------------------------
// Kernel 1: alpha (54), o1 (44226), o2 (44226)
// ---------------------------------------------------------------------------
__global__ void small_outputs_kernel(const float* __restrict__ sigma,
                                     const float* __restrict__ w,
                                     float* __restrict__ out) {
    int i = blockIdx.x * blockDim.x + threadIdx.x;
    if (i < ROWS_) {
        int g    = i / ROWNODE_;
        int rem  = i - g * ROWNODE_;
        int node = rem / 3;
        int l    = rem - node * 3;
        out[OFF_O1 + i] = sigma[(g * NODE_ + node) * 3 + l];
        out[OFF_O2 + i] = sigma[(g * NODE_ + id_map(node)) * 3 + l];
    }
    if (i < G_) {
        float a[3];
        softmax3(w, i, a);
        out[OFF_ALPHA + i * 3 + 0] = a[0];
        out[OFF_ALPHA + i * 3 + 1] = a[1];
        out[OFF_ALPHA + i * 3 + 2] = a[2];
    }
}

// ---------------------------------------------------------------------------
// Kernel 2: xn + lpn, one thread per element of (g,node,l0,q).
// lpn is forced to 0 for (bs>=1, node<729): skip all density work there.
// ---------------------------------------------------------------------------
__global__ void xn_lpn_kernel(const float* __restrict__ mu,
                              const float* __restrict__ sigma,
                              const float* __restrict__ w,
                              float* __restrict__ out) {
    int i = blockIdx.x * blockDim.x + threadIdx.x;
    if (i >= N_XN_) return;
    int q    = i % K_;
    int t    = i / K_;
    int l0   = t % 3;  t /= 3;
    int node = t % NODE_;
    int g    = t / NODE_;

    int base = (g * NODE_ + node) * 3;
    float mo = mu[base + l0];
    float so = sigma[base + l0];
    float xn = XQ[q] * so * SQRT2_F + mo;

    int  bs     = g % BS1_;
    bool masked = (bs >= 1) && (node < 729);
    float lpn = 0.0f;
    if (!masked) {
        float m0 = mu[base + 0], m1 = mu[base + 1], m2 = mu[base + 2];
        float s0 = sigma[base + 0], s1 = sigma[base + 1], s2 = sigma[base + 2];
        float a[3];
        softmax3(w, g, a);
        float i0 = 1.0f / s0, i1 = 1.0f / s1, i2 = 1.0f / s2;
        float t0 = (xn - m0) * i0, t1 = (xn - m1) * i1, t2 = (xn - m2) * i2;
        float s = a[0] * __expf(-0.5f * t0 * t0) * i0
                + a[1] * __expf(-0.5f * t1 * t1) * i1
                + a[2] * __expf(-0.5f * t2 * t2) * i2;
        lpn = __logf(s * INV_SQRT2PI_F + EPS_F);
    }

    __builtin_nontemporal_store(xn,  out + OFF_XN  + i);   // streamed, never re-read
    __builtin_nontemporal_store(lpn, out + OFF_LPN + i);
}

// ---------------------------------------------------------------------------
// Kernel 3: xe + lpe.  One block = 16 rows (row = (g,node,l0)); 4 waves split
// 18 dense column tiles of 16 quadrature points; column 288 is done scalar.
// The affine expansion x = (sqrt2*xq)*sigma + 1*mu is a K=2 GEMM done with
// V_WMMA_F32_16X16X4_F32.  The mixture sum uses the fused form
//   alpha*e1*e2 = coefF * exp(-0.5*(t1^2+t2^2))   (3 exps, not 6)
// ---------------------------------------------------------------------------

// Shared per-element epilogue: xe pair store + lpe density sum.
__device__ __forceinline__ void lpe_store(float* __restrict__ out,
                                          const float (*lds)[RSTRIDE],
                                          int M, int r, int q,
                                          float x1, float x2) {
    v2f pair; pair.x = x1; pair.y = x2;
    __builtin_nontemporal_store(pair,
        (v2f*)(out + OFF_XE + (size_t)r * (Q2_ * 2) + q * 2));

    float ssum = 0.0f;
    if (lds[M][RMSK] > 0.5f) {              // b1 == 1 region (wave-coherent)
#pragma unroll
        for (int l = 0; l < 3; ++l) {
            float t2 = (x2 - lds[M][RU2 + l]) * lds[M][RI2 + l];
            ssum += lds[M][RCM + l] * __expf(-0.5f * t2 * t2);
        }
    } else {
#pragma unroll
        for (int l = 0; l < 3; ++l) {
            float t1 = (x1 - lds[M][RU1 + l]) * lds[M][RI1 + l];
            float t2 = (x2 - lds[M][RU2 + l]) * lds[M][RI2 + l];
            ssum += lds[M][RCF + l] * __expf(-0.5f * (t1 * t1 + t2 * t2));
        }
    }
    __builtin_nontemporal_store(__logf(ssum + EPS_F),
                                out + OFF_LPE + (size_t)r * Q2_ + q);
}

template <bool GUARD>
__device__ __forceinline__ void emit_tile(float* __restrict__ out,
                                          const float (*lds)[RSTRIDE],
                                          int rowBase, int tile,
                                          int nloc, int halfsel,
                                          v8f d1, v8f d2) {
    int q = tile * 16 + nloc;                 // always < 289 (tiles 0..17)
#pragma unroll
    for (int i = 0; i < 8; ++i) {
        int M = i + (halfsel << 3);           // C/D layout: vgpr i, half sel
        int r = rowBase + M;
        if (!GUARD || r < ROWS_)
            lpe_store(out, lds, M, r, q, d1[i], d2[i]);
    }
}

__global__ __launch_bounds__(128) void xe_lpe_kernel(const float* __restrict__ mu,
                                                     const float* __restrict__ sigma,
                                                     const float* __restrict__ w,
                                                     float* __restrict__ out) {
    __shared__ float lds[16][RSTRIDE];

    const int rowBase = blockIdx.x * 16;
    const int tid     = threadIdx.x;

    if (tid < 16) {
        float vals[RSTRIDE];
#pragma unroll
        for (int j = 0; j < RSTRIDE; ++j) vals[j] = 0.0f;
        int r = rowBase + tid;
        if (r < ROWS_) {
            int g    = r / ROWNODE_;
            int rem  = r - g * ROWNODE_;
            int node = rem / 3;
            int l0   = rem - node * 3;
            int b1   = (g * NODE_ + node) * 3;
            int b2   = (g * NODE_ + id_map(node)) * 3;
            float a[3];
            softmax3(w, g, a);
#pragma unroll
            for (int l = 0; l < 3; ++l) {
                float u1 = mu[b1 + l], o1 = sigma[b1 + l];
                float u2 = mu[b2 + l], o2 = sigma[b2 + l];
                float i1 = 1.0f / o1, i2 = 1.0f / o2;
                vals[RU1 + l] = u1;  vals[RI1 + l] = i1;
                vals[RU2 + l] = u2;  vals[RI2 + l] = i2;
                vals[RCF + l] = a[l] * i1 * i2 * (INV_SQRT2PI_F * INV_SQRT2PI_F);
                vals[RCM + l] = a[l] * i2 * INV_SQRT2PI_F;
            }
            int bs     = g % BS1_;
            vals[RMSK] = (bs >= 1 && node < 729) ? 1.0f : 0.0f;
            vals[RA1X] = sigma[b1 + l0] * SQRT2_F;
            vals[RA1Y] = mu[b1 + l0];
            vals[RA2X] = sigma[b2 + l0] * SQRT2_F;
            vals[RA2Y] = mu[b2 + l0];
        }
#pragma unroll
        for (int j = 0; j < RSTRIDE; ++j) lds[tid][j] = vals[j];
    }
    __syncthreads();

    // ---- scalar tail: column q = 288 (xi = xj = XQ[16]) ----
    if (tid < 16) {
        int r = rowBase + tid;
        if (r < ROWS_) {
            float xk = XQ[16];
            float x1 = xk * lds[tid][RA1X] + lds[tid][RA1Y];
            float x2 = xk * lds[tid][RA2X] + lds[tid][RA2Y];
            lpe_store(out, lds, tid, r, Q2_ - 1, x1, x2);
        }
    }

    const int wave    = tid >> 5;
    const int lane    = tid & 31;
    const int halfsel = lane >> 4;   // 0: K=0,1 half, 1: K=2,3 half (zeros)
    const int nloc    = lane & 15;
    const bool tail   = (rowBase + 16 > ROWS_);   // true for exactly 1 block

    // A operands: 16x4 f32, lanes 0-15 hold M=lane {K0,K1}; lanes 16-31 -> 0
    v2f a1 = {0.0f, 0.0f}, a2 = {0.0f, 0.0f};
    if (halfsel == 0) {
        a1.x = lds[nloc][RA1X]; a1.y = lds[nloc][RA1Y];
        a2.x = lds[nloc][RA2X]; a2.y = lds[nloc][RA2Y];
    }

    for (int tile = wave; tile < 18; tile += 4) {  // wave-uniform: EXEC all-1s at WMMA
        int col = tile * 16 + nloc;                // <= 287, no clamp needed
        // B operands: 4x16 f32, lanes 0-15 hold N=lane {K0,K1}; lanes 16-31 -> 0
        v2f b1v = {0.0f, 0.0f}, b2v = {0.0f, 0.0f};
        if (halfsel == 0) {
            b1v.x = XQ[col % 17]; b1v.y = 1.0f;    // xi (fast index)
            b2v.x = XQ[col / 17]; b2v.y = 1.0f;    // xj (slow index)
        }
        v8f z = {};
        v8f d1 = __builtin_amdgcn_wmma_f32_16x16x4_f32(false, a1, false, b1v,
                                                       (short)0, z, false, false);
        v8f d2 = __builtin_amdgcn_wmma_f32_16x16x4_f32(false, a2, false, b2v,
                                                       (short)0, z, false, false);

        if (tail) emit_tile<true >(out, lds, rowBase, tile, nloc, halfsel, d1, d2);
        else      emit_tile<false>(out, lds, rowBase, tile, nloc, halfsel, d1, d2);
    }
}

// ---------------------------------------------------------------------------
extern "C" void kernel_launch(void* const* d_in, const int* in_sizes, int n_in,
                              void* d_out, int out_size, void* d_ws, size_t ws_size,
                              hipStream_t stream) {
    const float* mu    = (const float*)d_in[0];
    const float* sigma = (const float*)d_in[1];
    const float* w     = (const float*)d_in[2];
    float* out = (float*)d_out;

    small_outputs_kernel<<<(ROWS_ + 255) / 256, 256, 0, stream>>>(sigma, w, out);
    xn_lpn_kernel<<<(N_XN_ + 255) / 256, 256, 0, stream>>>(mu, sigma, w, out);
    xe_lpe_kernel<<<(ROWS_ + 15) / 16, 128, 0, stream>>>(mu, sigma, w, out);
}